// GRCU_GAT_75694503625339
// MI455X (gfx1250) — compile-verified
//
#include <hip/hip_runtime.h>

#define IN_F    64
#define OUT_F   64
#define HID     4096
#define NNODES  50000
#define NEDGES  800000
#define TSTEPS  3
#define NEG_SLOPE 0.2f

typedef float v2f __attribute__((ext_vector_type(2)));
typedef float v8f __attribute__((ext_vector_type(8)));

__device__ __forceinline__ float sigf(float x) { return 1.0f / (1.0f + expf(-x)); }

// ordered-uint mapping for float atomicMax (monotonic over all finite floats)
__device__ __forceinline__ unsigned fkey(float f) {
    unsigned u = __float_as_uint(f);
    return (u & 0x80000000u) ? ~u : (u | 0x80000000u);
}
__device__ __forceinline__ float keyf(unsigned k) {
    unsigned u = (k & 0x80000000u) ? (k ^ 0x80000000u) : ~k;
    return __uint_as_float(u);
}

// ---------------------------------------------------------------------------
// Kernel A: inp[f] = sum_n softmax(mask)[n] * x[n,f]   (single block, 1024 thr)
// ---------------------------------------------------------------------------
__global__ void softmax_pool_kernel(const float* __restrict__ mask,
                                    const float* __restrict__ x,
                                    float* __restrict__ inp) {
    __shared__ float red[1024];
    __shared__ float m_sh, z_sh;
    const int tid = threadIdx.x;

    float m = -3.0e38f;
    for (int n = tid; n < NNODES; n += 1024) m = fmaxf(m, mask[n]);
    red[tid] = m; __syncthreads();
    for (int s = 512; s > 0; s >>= 1) {
        if (tid < s) red[tid] = fmaxf(red[tid], red[tid + s]);
        __syncthreads();
    }
    if (tid == 0) m_sh = red[0];
    __syncthreads();
    m = m_sh;

    float z = 0.0f;
    for (int n = tid; n < NNODES; n += 1024) z += expf(mask[n] - m);
    red[tid] = z; __syncthreads();
    for (int s = 512; s > 0; s >>= 1) {
        if (tid < s) red[tid] += red[tid + s];
        __syncthreads();
    }
    if (tid == 0) z_sh = red[0];
    __syncthreads();
    z = z_sh;

    const int f   = tid & 63;
    const int grp = tid >> 6;          // 16 node groups
    float acc = 0.0f;
    for (int n = grp; n < NNODES; n += 16)
        acc += expf(mask[n] - m) * x[(size_t)n * IN_F + f];
    red[tid] = acc; __syncthreads();
    if (tid < IN_F) {
        float s = 0.0f;
        for (int g = 0; g < 16; ++g) s += red[g * 64 + f];
        inp[f] = s / z;
    }
}

// ---------------------------------------------------------------------------
// Kernel B: gates[r] = W_ih[r,:]·inp + b_ih[r] + W_hh[r,:]·h + b_hh[r]
// one wave32 per row; memory-bound GEMV over W_hh (256 MB/step)
// ---------------------------------------------------------------------------
__global__ void lstm_gates_kernel(const float* __restrict__ W_ih,
                                  const float* __restrict__ W_hh,
                                  const float* __restrict__ b_ih,
                                  const float* __restrict__ b_hh,
                                  const float* __restrict__ inp,
                                  const float* __restrict__ hstate,
                                  float* __restrict__ gates) {
    const int wave = threadIdx.x >> 5;
    const int lane = threadIdx.x & 31;
    const int r = blockIdx.x * 8 + wave;           // 4*HID rows, grid = 2048
    const float* wi = W_ih + (size_t)r * IN_F;
    const float* wh = W_hh + (size_t)r * HID;
    float s = 0.0f;
    for (int k = lane; k < IN_F; k += 32) s += wi[k] * inp[k];
    for (int k = lane; k < HID;  k += 32) s += wh[k] * hstate[k];
    for (int off = 16; off > 0; off >>= 1) s += __shfl_down(s, off, 32);
    if (lane == 0) gates[r] = s + b_ih[r] + b_hh[r];
}

// ---------------------------------------------------------------------------
// Kernel C: LSTM pointwise; h (=W_t flat) and c updated in place
// ---------------------------------------------------------------------------
__global__ void lstm_point_kernel(const float* __restrict__ gates,
                                  float* __restrict__ h, float* __restrict__ c) {
    const int u = blockIdx.x * 256 + threadIdx.x;
    if (u >= HID) return;
    const float gi = gates[u];
    const float gf = gates[HID + u];
    const float gg = gates[2 * HID + u];
    const float go = gates[3 * HID + u];
    const float c2 = sigf(gf) * c[u] + sigf(gi) * tanhf(gg);
    const float h2 = sigf(go) * tanhf(c2);
    c[u] = c2;
    h[u] = h2;
}

// ---------------------------------------------------------------------------
// Kernel D: hfeat = x @ W_t   ([50000,64]x[64,64]) via V_WMMA_F32_16X16X4_F32
// block = 256 thr = 8 waves; each wave computes a 16x64 tile.
// ---------------------------------------------------------------------------
__global__ void gemm_xW_wmma_kernel(const float* __restrict__ x,
                                    const float* __restrict__ Wt,
                                    float* __restrict__ hfeat, int nrows) {
    __shared__ float Ws[64 * 64];
    for (int i = threadIdx.x; i < 64 * 64; i += blockDim.x) Ws[i] = Wt[i];
    __syncthreads();

    const int wave = threadIdx.x >> 5;
    const int lane = threadIdx.x & 31;
    const int tile = blockIdx.x * 8 + wave;
    if (tile * 16 >= nrows) return;             // wave-uniform; EXEC stays full
    const int row0 = tile * 16;

    const int m    = lane & 15;                 // A: M index
    const int koff = (lane >> 4) << 1;          // A/B: K sub-offset (0 or 2)
    const int n0   = lane & 15;                 // B/D: N index within 16
    const float* xrow = x + (size_t)(row0 + m) * IN_F;

    v8f acc0 = {}, acc1 = {}, acc2 = {}, acc3 = {};
#pragma unroll
    for (int ks = 0; ks < 16; ++ks) {
        const int kb = ks * 4;
        v2f a;
        a.x = xrow[kb + koff];
        a.y = xrow[kb + koff + 1];
        const float* w0 = Ws + (kb + koff) * 64;     // row K
        const float* w1 = w0 + 64;                   // row K+1
        v2f b0, b1, b2, b3;
        b0.x = w0[n0];      b0.y = w1[n0];
        b1.x = w0[n0 + 16]; b1.y = w1[n0 + 16];
        b2.x = w0[n0 + 32]; b2.y = w1[n0 + 32];
        b3.x = w0[n0 + 48]; b3.y = w1[n0 + 48];
        acc0 = __builtin_amdgcn_wmma_f32_16x16x4_f32(false, a, false, b0, (short)0, acc0, false, false);
        acc1 = __builtin_amdgcn_wmma_f32_16x16x4_f32(false, a, false, b1, (short)0, acc1, false, false);
        acc2 = __builtin_amdgcn_wmma_f32_16x16x4_f32(false, a, false, b2, (short)0, acc2, false, false);
        acc3 = __builtin_amdgcn_wmma_f32_16x16x4_f32(false, a, false, b3, (short)0, acc3, false, false);
    }

    const int rhalf = (lane >> 4) << 3;          // D: M = r (+8 for upper lanes)
#pragma unroll
    for (int r = 0; r < 8; ++r) {
        float* orow = hfeat + (size_t)(row0 + rhalf + r) * OUT_F;
        orow[n0]      = acc0[r];
        orow[n0 + 16] = acc1[r];
        orow[n0 + 32] = acc2[r];
        orow[n0 + 48] = acc3[r];
    }
}

// ---------------------------------------------------------------------------
// Kernel E: per-node attention scores s_l = h·att_l, s_r = h·att_r
// ---------------------------------------------------------------------------
__global__ void att_scores_kernel(const float* __restrict__ hfeat,
                                  const float* __restrict__ att,
                                  float* __restrict__ sl, float* __restrict__ sr) {
    const int n = blockIdx.x * 256 + threadIdx.x;
    if (n >= NNODES) return;
    const float* hr = hfeat + (size_t)n * OUT_F;
    float a = 0.0f, b = 0.0f;
    for (int f = 0; f < OUT_F; ++f) {
        const float v = hr[f];
        a += v * att[f];
        b += v * att[OUT_F + f];
    }
    sl[n] = a;
    sr[n] = b;
}

// ---------------------------------------------------------------------------
// Kernel F: e = leaky_relu(s_l[src]+s_r[dst]); segment-max into mkey[dst]
// ---------------------------------------------------------------------------
__global__ void edge_score_kernel(const int* __restrict__ ei,
                                  const float* __restrict__ sl,
                                  const float* __restrict__ sr,
                                  float* __restrict__ ebuf,
                                  unsigned* __restrict__ mkey) {
    const int e = blockIdx.x * 256 + threadIdx.x;
    if (e >= NEDGES) return;
    const int src = ei[e];
    const int dst = ei[NEDGES + e];
    float v = sl[src] + sr[dst];
    v = (v > 0.0f) ? v : NEG_SLOPE * v;
    ebuf[e] = v;
    atomicMax(&mkey[dst], fkey(v));
}

// ---------------------------------------------------------------------------
// Kernel G: ex = exp(e - max[dst]); denom[dst] += ex
// ---------------------------------------------------------------------------
__global__ void edge_exp_kernel(const int* __restrict__ ei,
                                const float* __restrict__ ebuf,
                                const unsigned* __restrict__ mkey,
                                float* __restrict__ exbuf,
                                float* __restrict__ denom) {
    const int e = blockIdx.x * 256 + threadIdx.x;
    if (e >= NEDGES) return;
    const int dst = ei[NEDGES + e];
    const float ex = expf(ebuf[e] - keyf(mkey[dst]));
    exbuf[e] = ex;
    atomicAdd(&denom[dst], ex);
}

// ---------------------------------------------------------------------------
// Kernel H: out[dst,:] += (alpha*ew) * h[src,:]   one wave32 per edge
// ---------------------------------------------------------------------------
__global__ void edge_agg_kernel(const int* __restrict__ ei,
                                const float* __restrict__ ew,
                                const float* __restrict__ exbuf,
                                const float* __restrict__ denom,
                                const float* __restrict__ hfeat,
                                float* __restrict__ out) {
    const int wave = blockIdx.x * 8 + (threadIdx.x >> 5);
    const int lane = threadIdx.x & 31;
    if (wave >= NEDGES) return;
    const int src = ei[wave];
    const int dst = ei[NEDGES + wave];
    const float coef = exbuf[wave] / (denom[dst] + 1e-16f) * ew[wave];
    const float* hs = hfeat + (size_t)src * OUT_F;
    float* o = out + (size_t)dst * OUT_F;
    atomicAdd(&o[lane],      coef * hs[lane]);
    atomicAdd(&o[lane + 32], coef * hs[lane + 32]);
}

// ---------------------------------------------------------------------------
extern "C" void kernel_launch(void* const* d_in, const int* in_sizes, int n_in,
                              void* d_out, int out_size, void* d_ws, size_t ws_size,
                              hipStream_t stream) {
    const float* node_embs = (const float*)d_in[0];   // [T, N, 64]
    const float* mask      = (const float*)d_in[1];   // [T, N]
    const int*   edge_idx  = (const int*)  d_in[2];   // [T, 2, E]
    const float* edge_w    = (const float*)d_in[3];   // [T, E]
    const float* W_ih      = (const float*)d_in[4];   // [4H, 64]
    const float* W_hh      = (const float*)d_in[5];   // [4H, H]
    const float* b_ih      = (const float*)d_in[6];   // [4H]
    const float* b_hh      = (const float*)d_in[7];   // [4H]
    const float* W0        = (const float*)d_in[8];   // [64, 64]
    const float* att       = (const float*)d_in[9];   // [2, 64]
    float* out = (float*)d_out;                       // [T, N, 64]

    // workspace layout
    float*    hstate = (float*)d_ws;                  // HID
    float*    cstate = hstate + HID;                  // HID
    float*    gates  = cstate + HID;                  // 4*HID
    float*    inp    = gates + 4 * HID;               // 64
    float*    hfeat  = inp + 64;                      // N*64
    float*    sl     = hfeat + (size_t)NNODES * 64;   // N
    float*    sr     = sl + NNODES;                   // N
    unsigned* mkey   = (unsigned*)(sr + NNODES);      // N
    float*    denom  = (float*)(mkey + NNODES);       // N
    float*    ebuf   = denom + NNODES;                // E
    float*    exbuf  = ebuf + NEDGES;                 // E

    // init state: h = W0 flat, c = 0, out = 0
    hipMemcpyAsync(hstate, W0, HID * sizeof(float), hipMemcpyDeviceToDevice, stream);
    hipMemsetAsync(cstate, 0, HID * sizeof(float), stream);
    hipMemsetAsync(d_out, 0, (size_t)out_size * sizeof(float), stream);

    for (int t = 0; t < TSTEPS; ++t) {
        const float* x_t  = node_embs + (size_t)t * NNODES * IN_F;
        const float* m_t  = mask + (size_t)t * NNODES;
        const int*   ei_t = edge_idx + (size_t)t * 2 * NEDGES;
        const float* ew_t = edge_w + (size_t)t * NEDGES;
        float* out_t = out + (size_t)t * NNODES * OUT_F;

        hipMemsetAsync(mkey,  0, NNODES * sizeof(unsigned), stream);
        hipMemsetAsync(denom, 0, NNODES * sizeof(float), stream);

        softmax_pool_kernel<<<1, 1024, 0, stream>>>(m_t, x_t, inp);
        lstm_gates_kernel<<<(4 * HID) / 8, 256, 0, stream>>>(
            W_ih, W_hh, b_ih, b_hh, inp, hstate, gates);
        lstm_point_kernel<<<(HID + 255) / 256, 256, 0, stream>>>(gates, hstate, cstate);
        // hstate now holds W_t = h2.reshape(64,64)
        gemm_xW_wmma_kernel<<<(NNODES / 16 + 7) / 8, 256, 0, stream>>>(
            x_t, hstate, hfeat, NNODES);
        att_scores_kernel<<<(NNODES + 255) / 256, 256, 0, stream>>>(hfeat, att, sl, sr);
        edge_score_kernel<<<(NEDGES + 255) / 256, 256, 0, stream>>>(ei_t, sl, sr, ebuf, mkey);
        edge_exp_kernel<<<(NEDGES + 255) / 256, 256, 0, stream>>>(ei_t, ebuf, mkey, exbuf, denom);
        edge_agg_kernel<<<(NEDGES + 7) / 8, 256, 0, stream>>>(
            ei_t, ew_t, exbuf, denom, hfeat, out_t);
    }
}